// GCN_41781441855678
// MI455X (gfx1250) — compile-verified
//
#include <hip/hip_runtime.h>
#include <hip/hip_bf16.h>

typedef float v2f __attribute__((ext_vector_type(2)));
typedef float v8f __attribute__((ext_vector_type(8)));
typedef int   v4i __attribute__((ext_vector_type(4)));

#define NNODES 50000
#define EEDGES 800000
#define IN_DIM 256
#define HID    128
#define OUTD   16
#define BN_EPS 1e-5f

// ---- CDNA5 async Global->LDS staging (ASYNCcnt-tracked) -------------------
#if __has_builtin(__builtin_amdgcn_global_load_async_to_lds_b32)
#define HAVE_ASYNC_LDS 1
#else
#define HAVE_ASYNC_LDS 0
#endif

#if __has_builtin(__builtin_amdgcn_s_wait_asynccnt)
#define WAIT_ASYNC0() __builtin_amdgcn_s_wait_asynccnt(0)
#else
#define WAIT_ASYNC0() asm volatile("s_wait_asynccnt 0" ::: "memory")
#endif

// ---------------------------------------------------------------------------
// Degree / normalization
// ---------------------------------------------------------------------------
__global__ void init_deg_kernel(float* __restrict__ deg, int n) {
    int i = blockIdx.x * blockDim.x + threadIdx.x;
    if (i < n) deg[i] = 1.0f;  // self loop contributes 1
}

__global__ void count_deg_kernel(const long long* __restrict__ ei,
                                 float* __restrict__ deg, int ne) {
    int e = blockIdx.x * blockDim.x + threadIdx.x;
    if (e < ne) {
        int d = (int)ei[ne + e];  // row 1 = dst
        atomicAdd(&deg[d], 1.0f);
    }
}

__global__ void make_dinv_kernel(float* __restrict__ deg, int n) {
    int i = blockIdx.x * blockDim.x + threadIdx.x;
    if (i < n) deg[i] = rsqrtf(deg[i]);  // deg >= 1 always
}

// ---------------------------------------------------------------------------
// FP32 WMMA GEMM:  C[M x NC] = A[M x K] @ W[K x NC]   (K, NC compile-time)
// Block: 256 threads = 8 waves, each wave owns one 16-row M stripe and all
// NC/16 (<=8) column tiles. W K-slices are staged into LDS *transposed*
// (column-major, stride KSLICE+4) via per-lane async global->LDS b32 copies,
// so each B fragment pair is one conflict-free ds_load_b64.
// ---------------------------------------------------------------------------
#define KSLICE 64
#define LDS_STRIDE (KSLICE + 4)  // pad 4 floats -> disjoint banks per half-wave
#define WPB    8                 // waves per block

template <int K, int NC>
__global__ __launch_bounds__(256) void gemm_wmma_f32_kernel(
    const float* __restrict__ A, const float* __restrict__ W,
    float* __restrict__ C, int M) {
    __shared__ float ldsW[NC * LDS_STRIDE];  // 34 KB max (NC <= 128)

    const int tid   = threadIdx.x;
    const int wave  = tid >> 5;
    const int lane  = tid & 31;
    const int mrow  = lane & 15;  // M (for A) / N (for B) position in half-wave
    const int khalf = lane >> 4;  // which K pair this half-wave holds

    const int tile    = blockIdx.x * WPB + wave;
    const int m0      = tile * 16;
    const bool active = (m0 < M);

    constexpr int NT = NC / 16;  // 16-wide N tiles (8 or 1), compile-time
    v8f acc[NT];
#pragma unroll
    for (int t = 0; t < NT; ++t) acc[t] = (v8f){0, 0, 0, 0, 0, 0, 0, 0};

    const float* Arow = A + (size_t)(active ? (m0 + mrow) : mrow) * K;
    // per-lane base into the transposed tile: column (t*16 + mrow), row 2*khalf
    float* ldsB = ldsW + mrow * LDS_STRIDE + 2 * khalf;

    for (int kk = 0; kk < K; kk += KSLICE) {
        __syncthreads();
        // stage W[kk .. kk+KSLICE) x NC into LDS, transposed:
        //   ldsW[n * LDS_STRIDE + k] = W[kk + k][n]
        constexpr int total = KSLICE * NC;
        const float* Wsl = W + (size_t)kk * NC;
#pragma unroll
        for (int i = tid; i < total; i += 256) {
            const int k = i / NC;  // compile-time strength-reduced (NC pow2)
            const int n = i & (NC - 1);
#if HAVE_ASYNC_LDS
            __builtin_amdgcn_global_load_async_to_lds_b32(
                (int*)(Wsl + i), (int*)(ldsW + n * LDS_STRIDE + k),
                /*imm offset*/ 0, /*cpol*/ 0);
#else
            ldsW[n * LDS_STRIDE + k] = Wsl[i];
#endif
        }
#if HAVE_ASYNC_LDS
        WAIT_ASYNC0();
#endif
        __syncthreads();

#pragma unroll 4
        for (int kl = 0; kl < KSLICE; kl += 4) {
            // A fragment: 16x4 fp32 -> 2 VGPRs per lane (global b64)
            v2f a = *reinterpret_cast<const v2f*>(Arow + kk + kl + 2 * khalf);
#pragma unroll
            for (int t = 0; t < NT; ++t) {
                // B fragment pair in one ds_load_b64 (adjacent, 8B aligned)
                v2f b = *reinterpret_cast<const v2f*>(ldsB + t * 16 * LDS_STRIDE + kl);
                acc[t] = __builtin_amdgcn_wmma_f32_16x16x4_f32(
                    false, a, false, b, (short)0, acc[t], false, false);
            }
        }
    }

    if (active) {
        const int col = mrow;
#pragma unroll
        for (int t = 0; t < NT; ++t) {
#pragma unroll
            for (int r = 0; r < 8; ++r) {
                C[(size_t)(m0 + r + 8 * khalf) * NC + t * 16 + col] = acc[t][r];
            }
        }
    }
}

// ---------------------------------------------------------------------------
// Aggregation: out[i][f] = h[i][f]*dinv[i]^2 + bias[f]  (self-loop + bias)
// then edge scatter: out[dst][f] += h[src][f] * dinv[src]*dinv[dst]
// ---------------------------------------------------------------------------
template <int F>
__global__ void self_init_kernel(const float* __restrict__ h,
                                 const float* __restrict__ dinv,
                                 const float* __restrict__ bias,
                                 float* __restrict__ out, int n) {
    int idx = blockIdx.x * blockDim.x + threadIdx.x;
    if (idx < n * F) {
        int i = idx / F;
        int f = idx & (F - 1);
        float d = dinv[i];
        out[idx] = h[idx] * d * d + bias[f];
    }
}

template <int F>
__global__ void edge_scatter_kernel(const long long* __restrict__ ei,
                                    const float* __restrict__ h,
                                    const float* __restrict__ dinv,
                                    float* __restrict__ out, int ne) {
    constexpr int PER = 256 / F;  // edges per block (F=128 -> 2, F=16 -> 16)
    int e = blockIdx.x * PER + threadIdx.x / F;
    int f = threadIdx.x & (F - 1);
    if (e < ne) {
        int s = (int)ei[e];
        int d = (int)ei[ne + e];
        float nrm = dinv[s] * dinv[d];
        atomicAdd(&out[(size_t)d * F + f], h[(size_t)s * F + f] * nrm);
    }
}

// ---------------------------------------------------------------------------
// BatchNorm (training stats) + affine + ReLU (in place)
// ---------------------------------------------------------------------------
__global__ void bn_zero_kernel(float* __restrict__ sums) {
    int i = blockIdx.x * blockDim.x + threadIdx.x;
    if (i < 2 * HID) sums[i] = 0.0f;
}

#define BN_RPB 256  // rows per block
__global__ __launch_bounds__(HID) void bn_stats_kernel(
    const float* __restrict__ h, float* __restrict__ sums, int n) {
    int f  = threadIdx.x;  // blockDim.x == HID
    int r0 = blockIdx.x * BN_RPB;
    int r1 = min(r0 + BN_RPB, n);
    float s = 0.0f, s2 = 0.0f;
    for (int r = r0; r < r1; ++r) {
        float v = h[(size_t)r * HID + f];
        s += v;
        s2 += v * v;
    }
    atomicAdd(&sums[f], s);
    atomicAdd(&sums[HID + f], s2);
}

__global__ void bn_apply_relu_kernel(const float* __restrict__ sums,
                                     const float* __restrict__ g,
                                     const float* __restrict__ be,
                                     float* __restrict__ h, int n) {
    int idx = blockIdx.x * blockDim.x + threadIdx.x;
    if (idx < n * HID) {
        int f = idx & (HID - 1);
        float inv_n = 1.0f / (float)n;
        float m = sums[f] * inv_n;
        float v = sums[HID + f] * inv_n - m * m;
        float x = (h[idx] - m) * rsqrtf(v + BN_EPS) * g[f] + be[f];
        h[idx] = fmaxf(x, 0.0f);
    }
}

// ---------------------------------------------------------------------------
// Launcher
// ---------------------------------------------------------------------------
extern "C" void kernel_launch(void* const* d_in, const int* in_sizes, int n_in,
                              void* d_out, int out_size, void* d_ws, size_t ws_size,
                              hipStream_t stream) {
    const float*     x   = (const float*)d_in[0];
    const long long* ei  = (const long long*)d_in[1];  // int64 [2, E]
    const float*     W1  = (const float*)d_in[2];
    const float*     b1  = (const float*)d_in[3];
    const float*     g1  = (const float*)d_in[4];
    const float*     be1 = (const float*)d_in[5];
    const float*     W2  = (const float*)d_in[6];
    const float*     b2  = (const float*)d_in[7];
    const float*     g2  = (const float*)d_in[8];
    const float*     be2 = (const float*)d_in[9];
    const float*     W3  = (const float*)d_in[10];
    const float*     b3  = (const float*)d_in[11];
    float*           out = (float*)d_out;

    const int n = NNODES, ne = EEDGES;

    // workspace layout (floats)
    float* ws   = (float*)d_ws;
    float* dinv = ws;                      // n
    float* sums = ws + n;                  // 2*HID
    float* Hlin = sums + 2 * HID;          // n*HID
    float* Hagg = Hlin + (size_t)n * HID;  // n*HID

    const int mtiles  = n / 16;                    // 3125
    const int gblocks = (mtiles + WPB - 1) / WPB;  // 391

    // --- degree / normalization -------------------------------------------
    init_deg_kernel<<<(n + 255) / 256, 256, 0, stream>>>(dinv, n);
    count_deg_kernel<<<(ne + 255) / 256, 256, 0, stream>>>(ei, dinv, ne);
    make_dinv_kernel<<<(n + 255) / 256, 256, 0, stream>>>(dinv, n);

    // --- layer 1: conv(x, W1, b1) -> BN -> ReLU ---------------------------
    gemm_wmma_f32_kernel<IN_DIM, HID><<<gblocks, 256, 0, stream>>>(x, W1, Hlin, n);
    self_init_kernel<HID><<<(n * HID + 255) / 256, 256, 0, stream>>>(Hlin, dinv, b1, Hagg, n);
    edge_scatter_kernel<HID><<<(ne + 1) / 2, 256, 0, stream>>>(ei, Hlin, dinv, Hagg, ne);
    bn_zero_kernel<<<1, 256, 0, stream>>>(sums);
    bn_stats_kernel<<<(n + BN_RPB - 1) / BN_RPB, HID, 0, stream>>>(Hagg, sums, n);
    bn_apply_relu_kernel<<<(n * HID + 255) / 256, 256, 0, stream>>>(sums, g1, be1, Hagg, n);

    // --- layer 2: conv(h, W2, b2) -> BN -> ReLU ---------------------------
    gemm_wmma_f32_kernel<HID, HID><<<gblocks, 256, 0, stream>>>(Hagg, W2, Hlin, n);
    self_init_kernel<HID><<<(n * HID + 255) / 256, 256, 0, stream>>>(Hlin, dinv, b2, Hagg, n);
    edge_scatter_kernel<HID><<<(ne + 1) / 2, 256, 0, stream>>>(ei, Hlin, dinv, Hagg, ne);
    bn_zero_kernel<<<1, 256, 0, stream>>>(sums);
    bn_stats_kernel<<<(n + BN_RPB - 1) / BN_RPB, HID, 0, stream>>>(Hagg, sums, n);
    bn_apply_relu_kernel<<<(n * HID + 255) / 256, 256, 0, stream>>>(sums, g2, be2, Hagg, n);

    // --- layer 3: conv(h, W3, b3) -> d_out --------------------------------
    gemm_wmma_f32_kernel<HID, OUTD><<<gblocks, 256, 0, stream>>>(Hagg, W3, Hlin, n);
    self_init_kernel<OUTD><<<(n * OUTD + 255) / 256, 256, 0, stream>>>(Hlin, dinv, b3, out, n);
    edge_scatter_kernel<OUTD><<<(ne + 15) / 16, 256, 0, stream>>>(ei, Hlin, dinv, out, ne);
}